// GraphAttentionLayer_76416058130706
// MI455X (gfx1250) — compile-verified
//
#include <hip/hip_runtime.h>
#include <math.h>

typedef float v2f __attribute__((ext_vector_type(2)));
typedef float v8f __attribute__((ext_vector_type(8)));
typedef unsigned int u32x4 __attribute__((ext_vector_type(4)));
typedef int i32x4 __attribute__((ext_vector_type(4)));
typedef int i32x8 __attribute__((ext_vector_type(8)));

#define IN_F   256
#define NHEAD  8
#define NDIM   256              // N_HIDDEN
#define NNODE  1024
#define NBATCH 8
#define WCOLS  (NHEAD * NDIM)   // 2048
#define NEGINF (-9e16f)
#define SLOPE  0.2f

#define PJ   128                // attention j-chunk staged as P tile
#define PJP  (PJ + 2)           // padded row (130 dwords -> bank-conflict free)
#define ADJP (NNODE + 4)        // padded adj row (bytes)

__device__ __forceinline__ v8f wmma_f32(v2f a, v2f b, v8f c) {
  // D = A(16x4,f32) * B(4x16,f32) + C(16x16,f32)
  return __builtin_amdgcn_wmma_f32_16x16x4_f32(false, a, false, b,
                                               (short)0, c, false, false);
}

// ---------------------------------------------------------------------------
// TDM: 1-D tile of `dwords` f32 from global -> LDS. Issued by the calling
// wave only (EXEC ignored by TDM); waits tensorcnt itself; caller must barrier.
// ---------------------------------------------------------------------------
#if defined(__has_builtin) && __has_builtin(__builtin_amdgcn_tensor_load_to_lds)
#define HAVE_TDM 1
__device__ __forceinline__ void tdm_load_row_f32(const float* gsrc,
                                                 float* lds_dst,
                                                 unsigned dwords) {
  const unsigned long long ga = (unsigned long long)(const void*)gsrc;
  const unsigned lds_off = (unsigned)(unsigned long long)(const void*)lds_dst;
  u32x4 g0;
  g0.x = 1u;                                           // count=1, user mode
  g0.y = lds_off;                                      // [63:32]  lds_addr
  g0.z = (unsigned)ga;                                 // [95:64]  gaddr lo
  g0.w = (unsigned)((ga >> 32) & 0x1FFFFFFull) | (2u << 30); // gaddr hi|type=2
  i32x8 g1;
  g1.s0 = (int)(2u << 16);      // data_size=4B; no mask/flags
  g1.s1 = (int)(dwords << 16);  // tensor_dim0 low16 at bits [63:48]
  g1.s2 = (int)(1u << 16);      // tensor_dim0 hi=0, tensor_dim1=1
  g1.s3 = (int)(dwords << 16);  // tensor_dim1 hi=0, tile_dim0
  g1.s4 = 0;                    // tile_dim1/2 unused
  g1.s5 = (int)dwords;          // tensor_dim0_stride low32
  g1.s6 = 0;
  g1.s7 = 0;
  const i32x4 z4 = {0, 0, 0, 0};
#if __has_include(<hip/amd_detail/amd_gfx1250_TDM.h>)
  const i32x8 z8 = {0, 0, 0, 0, 0, 0, 0, 0};
  __builtin_amdgcn_tensor_load_to_lds(g0, g1, z4, z4, z8, 0);   // 6-arg lane
#else
  __builtin_amdgcn_tensor_load_to_lds(g0, g1, z4, z4, 0);       // 5-arg lane
#endif
  __builtin_amdgcn_s_wait_tensorcnt(0);
}
#else
#define HAVE_TDM 0
#endif

// ---------------------------------------------------------------------------
// Kernel 1: ht[b,h,n,d] = (h @ W) reshaped.  One wave -> 16x32 C tile
// (A-fragment reused across two B tiles, 2 WMMAs / K-step).
// ---------------------------------------------------------------------------
__global__ void __launch_bounds__(256)
k_gemm_ht(const float* __restrict__ hin,
          const float* __restrict__ W,
          float* __restrict__ ht) {
  const int lane  = threadIdx.x & 31;
  const int warp  = threadIdx.x >> 5;
  const int tile  = blockIdx.x * 8 + warp;     // 32768 tiles total
  const int ctile = tile & 63;                 // 2048/32 col tiles
  const int rtile = tile >> 6;                 // 8192/16 row tiles
  const int i0 = rtile * 16;
  const int j0 = ctile * 32;
  const int mn  = lane & 15;
  const int grp = lane >> 4;
  const int k0  = grp << 1;

  const float* arow  = hin + (size_t)(i0 + mn) * IN_F;
  const float* bcol0 = W + j0 + mn;
  const float* bcol1 = W + j0 + 16 + mn;

  v8f acc0 = {}, acc1 = {};
#pragma unroll 4
  for (int k = 0; k < IN_F; k += 4) {
    const size_t r0 = (size_t)(k + k0) * WCOLS;
    const size_t r1 = (size_t)(k + k0 + 1) * WCOLS;
    v2f a = *(const v2f*)(arow + k + k0);
    v2f b0; b0.x = bcol0[r0]; b0.y = bcol0[r1];
    v2f b1; b1.x = bcol1[r0]; b1.y = bcol1[r1];
    acc0 = wmma_f32(a, b0, acc0);
    acc1 = wmma_f32(a, b1, acc1);
  }

  const int head = (j0 + mn) >> 8;             // both 16-col tiles same head
  const int d0_  = (j0 + mn) & 255;
  const int d1_  = (j0 + 16 + mn) & 255;
#pragma unroll
  for (int r = 0; r < 8; ++r) {
    const int row = i0 + r + (grp << 3);
    const int b_  = row >> 10;
    const int n   = row & 1023;
    const size_t base = ((size_t)(b_ * NHEAD + head) * NNODE + n) * NDIM;
    ht[base + d0_] = acc0[r];
    ht[base + d1_] = acc1[r];
  }
}

// ---------------------------------------------------------------------------
// Kernel 2: src/tgt projections: src[bhn] = ht[bhn,:] . a_src[h], same for tgt
// ---------------------------------------------------------------------------
__global__ void __launch_bounds__(256)
k_src_tgt(const float* __restrict__ ht,
          const float* __restrict__ a,
          float* __restrict__ src,
          float* __restrict__ tgt) {
  const int i  = blockIdx.x * blockDim.x + threadIdx.x;   // over B*H*N
  const int hd = (i >> 10) & 7;
  const float* v  = ht + (size_t)i * NDIM;
  const float* as = a + (size_t)hd * (2 * NDIM);
  const float* at = as + NDIM;
  float s = 0.f, t = 0.f;
#pragma unroll 4
  for (int d = 0; d < NDIM; ++d) {
    const float x = v[d];
    s += x * as[d];
    t += x * at[d];
  }
  src[i] = s;
  tgt[i] = t;
}

// ---------------------------------------------------------------------------
// Kernel 3: per-row softmax stats (max, sum).  One wave per (b,h,i) row.
// ---------------------------------------------------------------------------
__global__ void __launch_bounds__(256)
k_softmax_stats(const float* __restrict__ src,
                const float* __restrict__ tgt,
                const int* __restrict__ adj,
                float* __restrict__ rowmax,
                float* __restrict__ rowsum) {
  const int lane = threadIdx.x & 31;
  const int warp = threadIdx.x >> 5;
  const int row  = blockIdx.x * 8 + warp;
  const int node = row & 1023;
  const int bh   = row >> 10;
  const float srow = src[row];
  const float* tg  = tgt + (size_t)bh * NNODE;
  const int* arow  = adj + (size_t)node * NNODE;

  float m = -INFINITY;
  for (int j = lane; j < NNODE; j += 32) {
    float e = srow + tg[j];
    e = (e >= 0.f) ? e : SLOPE * e;
    e = (arow[j] > 0) ? e : NEGINF;
    m = fmaxf(m, e);
  }
#pragma unroll
  for (int o = 16; o > 0; o >>= 1) m = fmaxf(m, __shfl_xor(m, o, 32));

  float s = 0.f;
  for (int j = lane; j < NNODE; j += 32) {
    float e = srow + tg[j];
    e = (e >= 0.f) ? e : SLOPE * e;
    e = (arow[j] > 0) ? e : NEGINF;
    s += __expf(e - m);
  }
#pragma unroll
  for (int o = 16; o > 0; o >>= 1) s += __shfl_xor(s, o, 32);

  if (lane == 0) { rowmax[row] = m; rowsum[row] = s; }
}

// ---------------------------------------------------------------------------
// Kernel 4: out[b,i,:] = mean_h( softmax(e) @ ht[b,h] ).
// One block (256 thr / 8 waves) per (b, 16-row i-tile); each wave owns a
// 16x32 output tile (2 C accumulators).  Each 16x128 P chunk is built ONCE
// per block into LDS (A-fragment layout, padded rows), then all waves stream
// it through V_WMMA_F32_16X16X4_F32.  tgt row staged per head via TDM.
// ---------------------------------------------------------------------------
__global__ void __launch_bounds__(256)
k_attn(const float* __restrict__ ht,
       const int* __restrict__ adj,
       const float* __restrict__ src,
       const float* __restrict__ tgt,
       const float* __restrict__ rowmax,
       const float* __restrict__ rowsum,
       float* __restrict__ out) {
  __shared__ unsigned char s_adj[16 * ADJP];   // ~16 KB
  __shared__ float s_tgt[NNODE];               // 4 KB
  __shared__ float s_P[16 * PJP];              // ~8 KB, padded rows
  __shared__ float s_src[16], s_m[16], s_rs[16];

  const int tid   = threadIdx.x;
  const int lane  = tid & 31;
  const int warp  = tid >> 5;                  // 0..7 -> 32-wide d tile
  const int b     = blockIdx.x >> 6;
  const int itile = blockIdx.x & 63;
  const int i0  = itile * 16;
  const int mn  = lane & 15;
  const int grp = lane >> 4;
  const int k0  = grp << 1;
  const int d0  = warp * 32;

  // Stage the 16 x 1024 adjacency strip once (shared by all 8 heads).
  for (int idx = tid; idx < 16 * NNODE; idx += 256) {
    const int mm = idx >> 10, j = idx & 1023;
    s_adj[mm * ADJP + j] = (unsigned char)(adj[(size_t)(i0 + mm) * NNODE + j] > 0);
  }

  v8f acc0 = {}, acc1 = {};
  for (int hd = 0; hd < NHEAD; ++hd) {
    const int bh = b * NHEAD + hd;
    __syncthreads();                           // adj ready / prev head drained
#if HAVE_TDM
    if (warp == 0)                             // one wave drives the TDM
      tdm_load_row_f32(tgt + (size_t)bh * NNODE, s_tgt, NNODE);
#else
    for (int j = tid; j < NNODE; j += 256)
      s_tgt[j] = tgt[(size_t)bh * NNODE + j];
#endif
    if (tid < 16) {
      const int rowg = bh * NNODE + i0 + tid;
      s_src[tid] = src[rowg];
      s_m[tid]   = rowmax[rowg];
      s_rs[tid]  = 1.0f / rowsum[rowg];
    }
    __syncthreads();

    const float* hv = ht + (size_t)bh * NNODE * NDIM + d0 + mn;  // column base

    for (int jc = 0; jc < NNODE; jc += PJ) {
      // Build one 16 x 128 P chunk cooperatively (2048 exps for all 8 waves).
      for (int idx = tid; idx < 16 * PJ; idx += 256) {
        const int mm = idx >> 7, jo = idx & (PJ - 1);
        const int j  = jc + jo;
        float e = s_src[mm] + s_tgt[j];
        e = (e >= 0.f) ? e : SLOPE * e;
        e = s_adj[mm * ADJP + j] ? e : NEGINF;
        s_P[mm * PJP + jo] = __expf(e - s_m[mm]) * s_rs[mm];
      }
      __syncthreads();

      const float* prow = s_P + mn * PJP;
#pragma unroll 2
      for (int jj = 0; jj < PJ; jj += 4) {
        const int j = jc + jj + k0;
        v2f A = *(const v2f*)(prow + jj + k0);          // ds_load_b64
        const size_t r0 = (size_t)j * NDIM;
        const size_t r1 = (size_t)(j + 1) * NDIM;
        v2f B0; B0.x = hv[r0];      B0.y = hv[r1];
        v2f B1; B1.x = hv[r0 + 16]; B1.y = hv[r1 + 16];
        acc0 = wmma_f32(A, B0, acc0);
        acc1 = wmma_f32(A, B1, acc1);
      }
      __syncthreads();                         // before s_P is overwritten
    }
  }

  // Head-mean (x 1/8) and store; each (b,n,d) written by exactly one lane.
#pragma unroll
  for (int r = 0; r < 8; ++r) {
    const int row = i0 + r + (grp << 3);
    const size_t base = ((size_t)(b * NNODE + row) * NDIM) + d0 + mn;
    out[base]      = acc0[r] * 0.125f;
    out[base + 16] = acc1[r] * 0.125f;
  }
}

// ---------------------------------------------------------------------------
extern "C" void kernel_launch(void* const* d_in, const int* in_sizes, int n_in,
                              void* d_out, int out_size, void* d_ws, size_t ws_size,
                              hipStream_t stream) {
  (void)in_sizes; (void)n_in; (void)out_size; (void)ws_size;
  const float* h   = (const float*)d_in[0];
  const int*   adj = (const int*)d_in[1];
  const float* W   = (const float*)d_in[2];
  const float* a   = (const float*)d_in[3];
  float* out = (float*)d_out;

  float* ws   = (float*)d_ws;
  float* ht   = ws;                                         // 16,777,216 f32
  float* src  = ht + (size_t)NBATCH * NHEAD * NNODE * NDIM; // 65536 f32
  float* tgt  = src + (size_t)NBATCH * NHEAD * NNODE;
  float* rmax = tgt + (size_t)NBATCH * NHEAD * NNODE;
  float* rsum = rmax + (size_t)NBATCH * NHEAD * NNODE;

  k_gemm_ht<<<4096, 256, 0, stream>>>(h, W, ht);
  k_src_tgt<<<256, 256, 0, stream>>>(ht, a, src, tgt);
  k_softmax_stats<<<8192, 256, 0, stream>>>(src, tgt, adj, rmax, rsum);
  k_attn<<<512, 256, 0, stream>>>(ht, adj, src, tgt, rmax, rsum, out);
}